// HistogramMLP_50053548867780
// MI455X (gfx1250) — compile-verified
//
#include <hip/hip_runtime.h>
#include <hip/hip_bf16.h>

typedef float v2f __attribute__((ext_vector_type(2)));
typedef float v8f __attribute__((ext_vector_type(8)));

#define HIST_BINS   64
#define N_PLANES    192          // 64 images * 3 channels
#define PLANE_PIX   (512 * 512)  // 262144
#define PLANE_F4    (PLANE_PIX / 4)       // 65536 float4 per plane
#define BLK_PER_PLANE 16
#define F4_PER_BLK  (PLANE_F4 / BLK_PER_PLANE)  // 4096
#define HIST_THREADS 256
#define ITERS       (F4_PER_BLK / HIST_THREADS) // 16
#define NWAVES      (HIST_THREADS / 32)         // 8

// ---------------- zero the count buffer ----------------
__global__ void zero_u32_kernel(unsigned int* __restrict__ p, int n) {
    int i = blockIdx.x * blockDim.x + threadIdx.x;
    if (i < n) p[i] = 0u;
}

// ---------------- histogram: 50.3M pixels, LDS-privatized bins ----------------
__global__ void hist_kernel(const float4* __restrict__ xf4,
                            unsigned int* __restrict__ ghist) {
    __shared__ unsigned int lh[NWAVES][HIST_BINS];
    const int tid  = threadIdx.x;
    const int wave = tid >> 5;

    // zero LDS sub-histograms
    for (int i = tid; i < NWAVES * HIST_BINS; i += HIST_THREADS)
        ((unsigned int*)lh)[i] = 0u;
    __syncthreads();

    const int plane = blockIdx.x >> 4;        // / BLK_PER_PLANE
    const int bip   = blockIdx.x & (BLK_PER_PLANE - 1);
    const float4* base = xf4 + (size_t)plane * PLANE_F4 + (size_t)bip * F4_PER_BLK;
    unsigned int* mh = lh[wave];

    #pragma unroll 4
    for (int it = 0; it < ITERS; ++it) {
        float4 p = base[it * HIST_THREADS + tid];   // global_load_b128, coalesced
        int b0 = min(63, max(0, (int)(p.x * 64.0f)));
        int b1 = min(63, max(0, (int)(p.y * 64.0f)));
        int b2 = min(63, max(0, (int)(p.z * 64.0f)));
        int b3 = min(63, max(0, (int)(p.w * 64.0f)));
        atomicAdd(&mh[b0], 1u);
        atomicAdd(&mh[b1], 1u);
        atomicAdd(&mh[b2], 1u);
        atomicAdd(&mh[b3], 1u);
    }
    __syncthreads();

    // flush: 64 threads each reduce one bin across the 8 sub-histograms
    if (tid < HIST_BINS) {
        unsigned int s = 0;
        #pragma unroll
        for (int w = 0; w < NWAVES; ++w) s += lh[w][tid];
        if (s) atomicAdd(&ghist[plane * HIST_BINS + tid], s);
    }
}

// ---------------- per-image L1 normalization (row of 192 counts) ----------------
__global__ void finalize_kernel(const unsigned int* __restrict__ counts,
                                float* __restrict__ histf) {
    __shared__ float s[192];
    const int row = blockIdx.x;   // 0..63 (image)
    const int t   = threadIdx.x;  // 0..191 (3*64 features)
    float v = (float)counts[row * 192 + t];
    s[t] = v;
    __syncthreads();
    if (t < 64) s[t] = s[t] + s[t + 64] + s[t + 128];
    __syncthreads();
    if (t < 32) s[t] += s[t + 32];
    __syncthreads();
    if (t < 16) s[t] += s[t + 16];
    __syncthreads();
    if (t < 8)  s[t] += s[t + 8];
    __syncthreads();
    if (t < 4)  s[t] += s[t + 4];
    __syncthreads();
    if (t < 2)  s[t] += s[t + 2];
    __syncthreads();
    if (t < 1)  s[0] += s[1];
    __syncthreads();
    const float inv = 1.0f / fmaxf(s[0], 1e-12f);
    histf[row * 192 + t] = v * inv;
}

// ---------------- fp32 WMMA GEMM layer: D = act(A[64xK] * B[KxN] + bias) ----------------
// one wave (32 threads) per 16x16 output tile, V_WMMA_F32_16X16X4_F32 accumulation.
// K, N, RELU are compile-time: straight-line fully-resolved loop, no scalar branching.
template <int K, int N, bool RELU>
__global__ void mlp_layer_kernel(const float* __restrict__ A,     // [64 x K] row-major
                                 const float* __restrict__ Bm,    // [K x N]  row-major
                                 const float* __restrict__ bias,  // [N]
                                 float* __restrict__ D) {         // [64 x N]
    const int lane = threadIdx.x & 31;
    const int h    = lane >> 4;   // lane half: selects K sub-pair
    const int l    = lane & 15;   // M index (A) / N index (B,C,D) within tile

    constexpr int nTilesN = N >> 4;
    const int tm = (blockIdx.x / nTilesN) << 4;
    const int tn = (blockIdx.x % nTilesN) << 4;

    // C/D layout: c[i] -> row tm + i + 8*h, col tn + l. Bias is per-column.
    v8f c;
    const float bv = bias[tn + l];
    #pragma unroll
    for (int i = 0; i < 8; ++i) c[i] = bv;

    // A (16x4 f32) layout: lanes 0-15 & 16-31 both cover M=0..15; K = k0 + 2*h + vgpr
    const float* arow = A + (size_t)(tm + l) * K + 2 * h;
    // B (4x16 f32) layout: lanes give N; K = k0 + 2*h + vgpr
    const float* bcol = Bm + (size_t)(2 * h) * N + tn + l;

    #pragma unroll 8
    for (int k0 = 0; k0 < K; k0 += 4) {
        v2f a, b;
        a.x = arow[k0 + 0];
        a.y = arow[k0 + 1];
        b.x = bcol[(size_t)k0 * N];
        b.y = bcol[(size_t)k0 * N + N];
        c = __builtin_amdgcn_wmma_f32_16x16x4_f32(
                /*neg_a=*/false, a, /*neg_b=*/false, b,
                /*c_mod=*/(short)0, c, /*reuse_a=*/false, /*reuse_b=*/false);
    }

    #pragma unroll
    for (int i = 0; i < 8; ++i) {
        float v = c[i];
        if (RELU) v = fmaxf(v, 0.0f);
        D[(size_t)(tm + i + 8 * h) * N + (tn + l)] = v;
    }
}

extern "C" void kernel_launch(void* const* d_in, const int* in_sizes, int n_in,
                              void* d_out, int out_size, void* d_ws, size_t ws_size,
                              hipStream_t stream) {
    const float* x  = (const float*)d_in[0];   // [64,3,512,512]
    const float* W1 = (const float*)d_in[1];   // [192,128]
    const float* b1 = (const float*)d_in[2];   // [128]
    const float* W2 = (const float*)d_in[3];   // [128,128]
    const float* b2 = (const float*)d_in[4];   // [128]
    float* out = (float*)d_out;                // [64,128]

    // workspace layout
    unsigned int* counts = (unsigned int*)d_ws;                           // 12288 u32
    float* histf  = (float*)((char*)d_ws + 12288 * sizeof(unsigned int)); // 12288 f32
    float* hidden = (float*)((char*)d_ws + 2 * 12288 * sizeof(unsigned int)); // 64*128 f32

    // 1) zero counts
    zero_u32_kernel<<<(12288 + 255) / 256, 256, 0, stream>>>(counts, 12288);

    // 2) histogram: 192 planes * 16 blocks each
    hist_kernel<<<N_PLANES * BLK_PER_PLANE, HIST_THREADS, 0, stream>>>(
        (const float4*)x, counts);

    // 3) L1 normalize per image (row of 192)
    finalize_kernel<<<64, 192, 0, stream>>>(counts, histf);

    // 4) layer 1: [64x192]@[192x128]+b1, relu -> hidden   (4*8 tiles, 1 wave each)
    mlp_layer_kernel<192, 128, true><<<32, 32, 0, stream>>>(histf, W1, b1, hidden);

    // 5) layer 2: [64x128]@[128x128]+b2 -> out
    mlp_layer_kernel<128, 128, false><<<32, 32, 0, stream>>>(hidden, W2, b2, out);
}